// PointMatching_2173253452330
// MI455X (gfx1250) — compile-verified
//
#include <hip/hip_runtime.h>
#include <stdint.h>

#define B_N   128
#define K_N   1024
#define KTOP  3
#define C_MAX (B_N * K_N * KTOP)          // 393216
#define LOG_THRESH (-2.9957322735539909f) // ln(0.05)

#define R_TILE 8
#define TILE_BYTES (R_TILE * K_N * 4)     // 32768 bytes per tile

// ---------------- helpers ----------------

__device__ __forceinline__ void top3_insert(float v, float& t0, float& t1, float& t2) {
    if (v > t2) {
        if (v > t0)      { t2 = t1; t1 = t0; t0 = v; }
        else if (v > t1) { t2 = t1; t1 = v; }
        else             { t2 = v; }
    }
}

// CDNA5 async global->LDS copy: 16 bytes per lane, tracked by ASYNCcnt.
__device__ __forceinline__ void async_copy_b128(uint32_t ldsoff, uint32_t goff, uint64_t sbase) {
    asm volatile("global_load_async_to_lds_b128 %0, %1, %2"
                 :
                 : "v"(ldsoff), "v"(goff), "s"(sbase)
                 : "memory");
}
__device__ __forceinline__ void wait_asynccnt_0() { asm volatile("s_wait_asynccnt 0" ::: "memory"); }
__device__ __forceinline__ void wait_asynccnt_2() { asm volatile("s_wait_asynccnt 2" ::: "memory"); }

// ---------------- kernel 1: per-column 3rd-largest via async-LDS double buffer ----------------

__global__ __launch_bounds__(1024) void col_kth_kernel(const float* __restrict__ score,
                                                       const int* __restrict__ rmask,
                                                       const int* __restrict__ cmask,
                                                       float* __restrict__ col_kth) {
    __shared__ float buf[2][R_TILE * K_N];   // 2 x 32 KB double buffer
    __shared__ int   smask[K_N];             // row masks of this batch

    const int tid = threadIdx.x;
    const int b   = blockIdx.x;
    const uint64_t gbase = (uint64_t)(uintptr_t)(score + ((size_t)b << 20));

    smask[tid] = rmask[(b << 10) + tid];

    const uint32_t lofs0 = (uint32_t)(uintptr_t)(&buf[0][0]) + (uint32_t)tid * 16u;
    const uint32_t lofs1 = (uint32_t)(uintptr_t)(&buf[1][0]) + (uint32_t)tid * 16u;

    // issue tile 0 into buffer 0 (per wave: 2 async instrs; block total 32 KB)
    {
        const uint32_t g0 = (uint32_t)tid * 16u;
        async_copy_b128(lofs0,           g0,           gbase);
        async_copy_b128(lofs0 + 16384u,  g0 + 16384u,  gbase);
    }

    float t0 = -__builtin_inff(), t1 = t0, t2 = t0;
    const int cvalid = cmask[(b << 10) + tid];

    const int NT = K_N / R_TILE;  // 128 tiles
    for (int t = 0; t < NT; ++t) {
        if (t + 1 < NT) {
            const uint32_t lo = ((t + 1) & 1) ? lofs1 : lofs0;
            const uint32_t g0 = (uint32_t)((t + 1) * TILE_BYTES) + (uint32_t)tid * 16u;
            async_copy_b128(lo,           g0,           gbase);
            async_copy_b128(lo + 16384u,  g0 + 16384u,  gbase);
            wait_asynccnt_2();   // tile t complete, tile t+1 in flight
        } else {
            wait_asynccnt_0();
        }
        __syncthreads();

        const float* bp = &buf[t & 1][0];
        const int rbase = t * R_TILE;
#pragma unroll
        for (int rr = 0; rr < R_TILE; ++rr) {
            float v = bp[rr * K_N + tid];
            if (!smask[rbase + rr]) v = -__builtin_inff();
            top3_insert(v, t0, t1, t2);
        }
        __syncthreads();  // all reads of buf[t&1] done before it is refilled at t+2
    }
    if (!cvalid) t2 = -__builtin_inff();
    col_kth[(b << 10) + tid] = t2;
}

// ---------------- kernel 2 (fused): per-row 3rd-largest + selected-count ----------------
// One wave per row. Row is loaded ONCE into registers (8 x float4 per lane), the
// top-3 butterfly leaves the row kth-value in every lane, then the count sweep
// runs entirely from registers (no second HBM pass).

__global__ __launch_bounds__(256) void row_count_kernel(const float* __restrict__ score,
                                                        const int* __restrict__ rmask,
                                                        const int* __restrict__ cmask,
                                                        const float* __restrict__ col_kth,
                                                        float* __restrict__ row_kth,
                                                        unsigned* __restrict__ counts) {
    const int lane = threadIdx.x & 31;
    const int wave = blockIdx.x * 8 + (threadIdx.x >> 5);   // global row id, [0, B*K)
    const int b    = wave >> 10;
    const float* rowp = score + ((size_t)wave << 10);
    const int*   cm   = cmask + (b << 10);
    const float* ck   = col_kth + (b << 10);
    const int    rm   = rmask[wave];

    if (!rm) {  // wave-uniform: row fully masked -> kth=-inf, count=0
        if (lane == 0) { row_kth[wave] = -__builtin_inff(); counts[wave] = 0u; }
        return;
    }

    float4   vals[8];
    unsigned pm = 0;  // packed column-mask bits, bit (j*4+e)
    float t0 = -__builtin_inff(), t1 = t0, t2 = t0;

#pragma unroll
    for (int j = 0; j < 8; ++j) {
        const int cb = j * 128 + lane * 4;
        float4 s4 = *reinterpret_cast<const float4*>(rowp + cb);
        int4   m4 = *reinterpret_cast<const int4*>(cm + cb);
        vals[j] = s4;
        pm |= (m4.x ? 1u : 0u) << (j * 4 + 0);
        pm |= (m4.y ? 1u : 0u) << (j * 4 + 1);
        pm |= (m4.z ? 1u : 0u) << (j * 4 + 2);
        pm |= (m4.w ? 1u : 0u) << (j * 4 + 3);
        top3_insert(m4.x ? s4.x : -__builtin_inff(), t0, t1, t2);
        top3_insert(m4.y ? s4.y : -__builtin_inff(), t0, t1, t2);
        top3_insert(m4.z ? s4.z : -__builtin_inff(), t0, t1, t2);
        top3_insert(m4.w ? s4.w : -__builtin_inff(), t0, t1, t2);
    }
    // butterfly top-3 merge: every lane ends with the global row top-3
    for (int off = 16; off > 0; off >>= 1) {
        float a0 = __shfl_xor(t0, off, 32);
        float a1 = __shfl_xor(t1, off, 32);
        float a2 = __shfl_xor(t2, off, 32);
        top3_insert(a0, t0, t1, t2);
        top3_insert(a1, t0, t1, t2);
        top3_insert(a2, t0, t1, t2);
    }
    const float rk = t2;

    unsigned cnt = 0;
#pragma unroll
    for (int j = 0; j < 8; ++j) {
        const int cb = j * 128 + lane * 4;
        float4 s4  = vals[j];
        float4 ck4 = *reinterpret_cast<const float4*>(ck + cb);  // L2/WGP$ resident
        bool p0 = ((pm >> (j * 4 + 0)) & 1u) && (s4.x > LOG_THRESH) && (s4.x >= rk) && (s4.x >= ck4.x);
        bool p1 = ((pm >> (j * 4 + 1)) & 1u) && (s4.y > LOG_THRESH) && (s4.y >= rk) && (s4.y >= ck4.y);
        bool p2 = ((pm >> (j * 4 + 2)) & 1u) && (s4.z > LOG_THRESH) && (s4.z >= rk) && (s4.z >= ck4.z);
        bool p3 = ((pm >> (j * 4 + 3)) & 1u) && (s4.w > LOG_THRESH) && (s4.w >= rk) && (s4.w >= ck4.w);
        cnt += __popc(__builtin_amdgcn_ballot_w32(p0));
        cnt += __popc(__builtin_amdgcn_ballot_w32(p1));
        cnt += __popc(__builtin_amdgcn_ballot_w32(p2));
        cnt += __popc(__builtin_amdgcn_ballot_w32(p3));
    }
    if (lane == 0) { row_kth[wave] = rk; counts[wave] = cnt; }
}

// ---------------- kernel 3: exclusive scan of 131072 counts (single block) ----------------

__global__ __launch_bounds__(1024) void scan_kernel(const unsigned* __restrict__ counts,
                                                    unsigned* __restrict__ offsets) {
    __shared__ unsigned sh[1024];
    const int tid = threadIdx.x;
    const int PER = (B_N * K_N) / 1024;   // 128
    const int base = tid * PER;

    unsigned sum = 0;
    for (int i = 0; i < PER; ++i) sum += counts[base + i];
    sh[tid] = sum;
    __syncthreads();
    for (int off = 1; off < 1024; off <<= 1) {
        unsigned v = (tid >= off) ? sh[tid - off] : 0u;
        __syncthreads();
        sh[tid] += v;
        __syncthreads();
    }
    unsigned run = sh[tid] - sum;  // exclusive prefix of this thread's chunk
    for (int i = 0; i < PER; ++i) { offsets[base + i] = run; run += counts[base + i]; }
}

// ---------------- kernel 4: zero-fill padded outputs ----------------

__global__ void zero_kernel(float* __restrict__ out, int n) {
    int i = blockIdx.x * blockDim.x + threadIdx.x;
    if (i < n) out[i] = 0.0f;
}

// ---------------- kernel 5: ordered compaction + gather (sparse: ~1% of rows non-empty) ----------------

__global__ __launch_bounds__(256) void emit_kernel(const float* __restrict__ score,
                                                   const int* __restrict__ rmask,
                                                   const int* __restrict__ cmask,
                                                   const float* __restrict__ row_kth,
                                                   const float* __restrict__ col_kth,
                                                   const unsigned* __restrict__ counts,
                                                   const unsigned* __restrict__ offsets,
                                                   const float* __restrict__ sp,
                                                   const float* __restrict__ tp,
                                                   const int* __restrict__ si,
                                                   const int* __restrict__ ti,
                                                   const float* __restrict__ gs,
                                                   float* __restrict__ out) {
    float* o_sp = out;
    float* o_tp = out + (size_t)C_MAX * 3;
    float* o_si = out + (size_t)C_MAX * 6;
    float* o_ti = out + (size_t)C_MAX * 7;
    float* o_sc = out + (size_t)C_MAX * 8;

    const int lane = threadIdx.x & 31;
    const int wave = blockIdx.x * 8 + (threadIdx.x >> 5);

    if (counts[wave] == 0u) return;   // wave-uniform early exit: skip the row entirely

    const int b    = wave >> 10;
    const float* rowp = score + ((size_t)wave << 10);
    const int*   cm   = cmask + (b << 10);
    const float* ck   = col_kth + (b << 10);
    const int    rm   = rmask[wave];
    const float  rk   = row_kth[wave];
    const float  g    = gs[b];

    const float sx = sp[wave * 3 + 0], sy = sp[wave * 3 + 1], sz = sp[wave * 3 + 2];
    const float sidx = (float)si[wave];

    unsigned base = offsets[wave];
    const unsigned lt = (1u << lane) - 1u;

    for (int j = 0; j < 8; ++j) {
        const int cb = j * 128 + lane * 4;
        float4 s4  = *reinterpret_cast<const float4*>(rowp + cb);
        int4   m4  = *reinterpret_cast<const int4*>(cm + cb);
        float4 ck4 = *reinterpret_cast<const float4*>(ck + cb);
        bool p0 = rm && m4.x && (s4.x > LOG_THRESH) && (s4.x >= rk) && (s4.x >= ck4.x);
        bool p1 = rm && m4.y && (s4.y > LOG_THRESH) && (s4.y >= rk) && (s4.y >= ck4.y);
        bool p2 = rm && m4.z && (s4.z > LOG_THRESH) && (s4.z >= rk) && (s4.z >= ck4.z);
        bool p3 = rm && m4.w && (s4.w > LOG_THRESH) && (s4.w >= rk) && (s4.w >= ck4.w);
        unsigned b0 = (unsigned)__builtin_amdgcn_ballot_w32(p0);
        unsigned b1 = (unsigned)__builtin_amdgcn_ballot_w32(p1);
        unsigned b2 = (unsigned)__builtin_amdgcn_ballot_w32(p2);
        unsigned b3 = (unsigned)__builtin_amdgcn_ballot_w32(p3);
        // lanes hold consecutive groups of 4 columns -> lane-major prefix == ascending c order
        unsigned p = base + __popc(b0 & lt) + __popc(b1 & lt) + __popc(b2 & lt) + __popc(b3 & lt);

        float  vv[4] = { s4.x, s4.y, s4.z, s4.w };
        bool   pr[4] = { p0, p1, p2, p3 };
#pragma unroll
        for (int e = 0; e < 4; ++e) {
            if (pr[e]) {
                if (p < (unsigned)C_MAX) {
                    const int c = cb + e;
                    const int tcol = (b << 10) + c;
                    o_sp[p * 3 + 0] = sx;
                    o_sp[p * 3 + 1] = sy;
                    o_sp[p * 3 + 2] = sz;
                    o_tp[p * 3 + 0] = tp[tcol * 3 + 0];
                    o_tp[p * 3 + 1] = tp[tcol * 3 + 1];
                    o_tp[p * 3 + 2] = tp[tcol * 3 + 2];
                    o_si[p] = sidx;
                    o_ti[p] = (float)ti[tcol];
                    o_sc[p] = expf(vv[e]) * g;
                }
                ++p;
            }
        }
        base += __popc(b0) + __popc(b1) + __popc(b2) + __popc(b3);
    }
}

// ---------------- launch ----------------

extern "C" void kernel_launch(void* const* d_in, const int* in_sizes, int n_in,
                              void* d_out, int out_size, void* d_ws, size_t ws_size,
                              hipStream_t stream) {
    const float* sp    = (const float*)d_in[0];  // src_knn_points  (B,K,3)
    const float* tp    = (const float*)d_in[1];  // tgt_knn_points  (B,K,3)
    const int*   rmask = (const int*)d_in[2];    // src_knn_masks   (B,K)
    const int*   cmask = (const int*)d_in[3];    // tgt_knn_masks   (B,K)
    const int*   si    = (const int*)d_in[4];    // src_knn_indices (B,K)
    const int*   ti    = (const int*)d_in[5];    // tgt_knn_indices (B,K)
    const float* score = (const float*)d_in[6];  // score_mat       (B,K,K)
    const float* gs    = (const float*)d_in[7];  // global_scores   (B,)
    float* out = (float*)d_out;

    char* ws = (char*)d_ws;
    float*    row_kth = (float*)(ws);
    float*    col_kth = (float*)(ws + (512u << 10));
    unsigned* counts  = (unsigned*)(ws + (1024u << 10));
    unsigned* offsets = (unsigned*)(ws + (1536u << 10));

    const int rows_blocks = (B_N * K_N) / 8;  // 256 threads = 8 waves, one wave per row

    col_kth_kernel<<<B_N, 1024, 0, stream>>>(score, rmask, cmask, col_kth);
    row_count_kernel<<<rows_blocks, 256, 0, stream>>>(score, rmask, cmask, col_kth,
                                                      row_kth, counts);
    scan_kernel<<<1, 1024, 0, stream>>>(counts, offsets);
    const int n_out = C_MAX * 9;
    zero_kernel<<<(n_out + 1023) / 1024, 1024, 0, stream>>>(out, n_out);
    emit_kernel<<<rows_blocks, 256, 0, stream>>>(score, rmask, cmask, row_kth, col_kth,
                                                 counts, offsets, sp, tp, si, ti, gs, out);
}